// FCOS_67216238183075
// MI455X (gfx1250) — compile-verified
//
#include <hip/hip_runtime.h>
#include <stdint.h>

// ---------------- problem constants ----------------
#define T_CAND   86016          // 256^2 + 128^2 + 64^2
#define NCLS     80
#define K_TOP    1000
#define NSEL     3000
#define SORTN    4096
#define NWORDS   94             // ceil(3000/32)
#define NMS_TH   0.6f
#define CONF_TH  0.05f

typedef uint32_t u32x4 __attribute__((ext_vector_type(4)));
typedef uint32_t u32x8 __attribute__((ext_vector_type(8)));

// ---------------- workspace layout (uint32 units) ----------------
enum : uint32_t {
  OFF_SCORE  = 0,                          // 86016 f32
  OFF_LABEL  = OFF_SCORE + T_CAND,         // 86016 u32
  OFF_H1     = OFF_LABEL + T_CAND,         // 3*2048
  OFF_H2     = OFF_H1 + 3 * 2048,          // 3*2048
  OFF_META   = OFF_H2 + 3 * 2048,          // lvl*8: [0]=B1 [1]=cGT1 [2]=B2 [3]=defTot
  OFF_CNT    = OFF_META + 24,              // lvl*2: [0]=cntDef [1]=cntTie
  OFF_SELCID = OFF_CNT + 8,                // 3000
  OFF_SELKEY = OFF_SELCID + NSEL,          // 3000
  OFF_SORT   = OFF_SELKEY + NSEL,          // 3000
  OFF_MASK   = OFF_SORT + NSEL,            // 3000*94 suppression bitmask
  OFF_END    = OFF_MASK + NSEL * NWORDS
};
#define ZERO_N (OFF_SELCID - OFF_H1)       // hist1+hist2+meta+cnt

// ---------------- level helpers ----------------
__device__ __forceinline__ int   lvl_of(int t)    { return (t < 65536) ? 0 : (t < 81920 ? 1 : 2); }
__device__ __forceinline__ int   lvl_base(int l)  { return l == 0 ? 0 : (l == 1 ? 65536 : 81920); }
__device__ __forceinline__ int   lvl_hw(int l)    { return l == 0 ? 65536 : (l == 1 ? 16384 : 4096); }
__device__ __forceinline__ int   lvl_w(int l)     { return 256 >> l; }
__device__ __forceinline__ float lvl_stride(int l){ return (float)(8 << l); }

// ---------------- CDNA5 async global->LDS (ASYNCcnt path) ----------------
__device__ __forceinline__ uint32_t lds_off(const void* p) {
  // LDS aperture: hardware truncates flat address to addr[31:0] == LDS byte offset.
  return (uint32_t)(uintptr_t)p;
}
__device__ __forceinline__ void async_ld_b32(uint32_t loff, const void* g) {
  asm volatile("global_load_async_to_lds_b32 %0, %1, off" :: "v"(loff), "v"(g) : "memory");
}
__device__ __forceinline__ void wait_async0() {
  asm volatile("s_wait_asynccnt 0x0" ::: "memory");
}

__device__ __forceinline__ float fsig(float x) { return 1.0f / (1.0f + __expf(-x)); }

// ---------------- K0: zero hist/meta/counters ----------------
__global__ void k_zero(uint32_t* __restrict__ ws) {
  int i = blockIdx.x * blockDim.x + threadIdx.x;
  if (i < (int)ZERO_N) ws[OFF_H1 + i] = 0u;
}

// ---------------- K1: streaming class-max + joint score ----------------
// argmax over classes == argmax over raw logits (sigmoid/sqrt monotone);
// 4 positions per thread -> b128 plane loads, 4x fewer VMEM issues for the
// same 27.5MB stream; 2 transcendentals per position only.
__global__ void k_decode(const float* __restrict__ c0, const float* __restrict__ c1,
                         const float* __restrict__ c2, const float* __restrict__ n0,
                         const float* __restrict__ n1, const float* __restrict__ n2,
                         float* __restrict__ score, uint32_t* __restrict__ label) {
  int q = blockIdx.x * blockDim.x + threadIdx.x;     // quad index
  if (q >= T_CAND / 4) return;
  int t0 = q * 4;
  int l = lvl_of(t0);
  const float* cls = (l == 0) ? c0 : (l == 1 ? c1 : c2);
  const float* ctn = (l == 0) ? n0 : (l == 1 ? n1 : n2);
  int p  = t0 - lvl_base(l);
  int HW = lvl_hw(l);
  float4 m = *(const float4*)(cls + p);              // class 0
  int ax = 0, ay = 0, az = 0, aw = 0;
#pragma unroll 4
  for (int c = 1; c < NCLS; ++c) {
    float4 v = *(const float4*)(cls + c * HW + p);   // coalesced b128
    if (v.x > m.x) { m.x = v.x; ax = c; }
    if (v.y > m.y) { m.y = v.y; ay = c; }
    if (v.z > m.z) { m.z = v.z; az = c; }
    if (v.w > m.w) { m.w = v.w; aw = c; }
  }
  float4 ct = *(const float4*)(ctn + p);
  float4 s;
  s.x = sqrtf(fsig(m.x) * fsig(ct.x));
  s.y = sqrtf(fsig(m.y) * fsig(ct.y));
  s.z = sqrtf(fsig(m.z) * fsig(ct.z));
  s.w = sqrtf(fsig(m.w) * fsig(ct.w));
  *(float4*)(score + t0) = s;
  uint4 lb = { (uint32_t)ax, (uint32_t)ay, (uint32_t)az, (uint32_t)aw };
  *(uint4*)(label + t0) = lb;
}

// ---------------- K2a/K3a/K2b/K3b/K4: two-pass radix top-1000 per level ---
__global__ void k_hist1(const float* __restrict__ score, uint32_t* __restrict__ h1) {
  int t = blockIdx.x * blockDim.x + threadIdx.x;
  if (t >= T_CAND) return;
  int l = lvl_of(t);
  uint32_t key = __float_as_uint(score[t]);
  atomicAdd(&h1[l * 2048 + (key >> 21)], 1u);
}
__global__ void k_sel1(const uint32_t* __restrict__ h1, uint32_t* __restrict__ meta) {
  int l = threadIdx.x;
  if (l >= 3) return;
  uint32_t acc = 0; int B = 0;
  for (int b = 2047; b >= 0; --b) {
    uint32_t h = h1[l * 2048 + b];
    if (acc + h >= K_TOP) { B = b; break; }
    acc += h;
  }
  meta[l * 8 + 0] = (uint32_t)B;
  meta[l * 8 + 1] = acc;                  // count strictly above bin B
}
__global__ void k_hist2(const float* __restrict__ score, const uint32_t* __restrict__ meta,
                        uint32_t* __restrict__ h2) {
  int t = blockIdx.x * blockDim.x + threadIdx.x;
  if (t >= T_CAND) return;
  int l = lvl_of(t);
  uint32_t key = __float_as_uint(score[t]);
  if ((key >> 21) == meta[l * 8 + 0])
    atomicAdd(&h2[l * 2048 + ((key >> 10) & 2047u)], 1u);
}
__global__ void k_sel2(const uint32_t* __restrict__ h2, uint32_t* __restrict__ meta) {
  int l = threadIdx.x;
  if (l >= 3) return;
  uint32_t cgt1 = meta[l * 8 + 1];
  uint32_t K2 = K_TOP - cgt1;
  uint32_t acc = 0; int B = 0;
  for (int b = 2047; b >= 0; --b) {
    uint32_t h = h2[l * 2048 + b];
    if (acc + h >= K2) { B = b; break; }
    acc += h;
  }
  meta[l * 8 + 2] = (uint32_t)B;
  meta[l * 8 + 3] = cgt1 + acc;           // defTot: definite selections
}
__global__ void k_compact(const float* __restrict__ score, const uint32_t* __restrict__ meta,
                          uint32_t* __restrict__ cnt, uint32_t* __restrict__ selCid,
                          uint32_t* __restrict__ selKey) {
  int t = blockIdx.x * blockDim.x + threadIdx.x;
  if (t >= T_CAND) return;
  int l = lvl_of(t);
  uint32_t key = __float_as_uint(score[t]);
  uint32_t b1 = key >> 21, B1 = meta[l * 8 + 0];
  int slot = -1;
  if (b1 > B1) {
    slot = (int)atomicAdd(&cnt[l * 2 + 0], 1u);
  } else if (b1 == B1) {
    uint32_t b2 = (key >> 10) & 2047u;
    uint32_t B2 = meta[l * 8 + 2], defTot = meta[l * 8 + 3];
    if (b2 > B2) {
      slot = (int)atomicAdd(&cnt[l * 2 + 0], 1u);
    } else if (b2 == B2) {
      uint32_t r = atomicAdd(&cnt[l * 2 + 1], 1u);
      if (defTot + r < K_TOP) slot = (int)(defTot + r);
    }
  }
  if (slot >= 0) {
    int o = l * K_TOP + slot;
    selCid[o] = (uint32_t)t;
    selKey[o] = key;
  }
}

// ---------------- K5: in-LDS bitonic sort of 3000 (padded to 4096) --------
__global__ __launch_bounds__(1024) void k_sort(const uint32_t* __restrict__ selKey,
                                               const uint32_t* __restrict__ selCid,
                                               uint32_t* __restrict__ sorted,
                                               float* __restrict__ out) {
  __shared__ uint32_t skey[SORTN];
  __shared__ uint32_t spay[SORTN];
  int tid = threadIdx.x;
  for (int i = tid; i < SORTN; i += 1024) {
    if (i < NSEL) {
      async_ld_b32(lds_off(&skey[i]), &selKey[i]);   // ASYNCcnt-tracked staging
      async_ld_b32(lds_off(&spay[i]), &selCid[i]);
    } else {
      skey[i] = 0u; spay[i] = 0u;                    // pad sinks to the tail
    }
  }
  wait_async0();
  __syncthreads();
  for (unsigned k = 2; k <= SORTN; k <<= 1) {
    for (unsigned j = k >> 1; j > 0; j >>= 1) {
      for (unsigned i = (unsigned)tid; i < SORTN; i += 1024) {
        unsigned x = i ^ j;
        if (x > i) {
          uint32_t a = skey[i], b = skey[x];
          bool sw = ((i & k) == 0u) ? (a < b) : (a > b);   // descending
          if (sw) {
            skey[i] = b; skey[x] = a;
            uint32_t pa = spay[i]; spay[i] = spay[x]; spay[x] = pa;
          }
        }
      }
      __syncthreads();
    }
  }
  for (int i = tid; i < NSEL; i += 1024) {
    sorted[i] = spay[i];
    out[12000 + i] = __uint_as_float(skey[i]);       // scores_s
  }
}

// ---------------- K6: box decode for sorted candidates --------------------
__global__ void k_boxes(const uint32_t* __restrict__ sorted, const uint32_t* __restrict__ label,
                        const float* __restrict__ r0, const float* __restrict__ r1,
                        const float* __restrict__ r2, const float* __restrict__ scales,
                        float* __restrict__ out) {
  int i = blockIdx.x * blockDim.x + threadIdx.x;
  if (i >= NSEL) return;
  int cid = (int)sorted[i];
  int l = lvl_of(cid);
  int p = cid - lvl_base(l);
  int HW = lvl_hw(l), W = lvl_w(l);
  float st = lvl_stride(l), sc = scales[l];
  const float* reg = (l == 0) ? r0 : (l == 1 ? r1 : r2);
  float ax = ((float)(p % W) + 0.5f) * st;
  float ay = ((float)(p / W) + 0.5f) * st;
  float dl = fmaxf(reg[p] * sc, 0.f) * st;
  float dt = fmaxf(reg[HW + p] * sc, 0.f) * st;
  float dr = fmaxf(reg[2 * HW + p] * sc, 0.f) * st;
  float db = fmaxf(reg[3 * HW + p] * sc, 0.f) * st;
  out[4 * i + 0] = ax - dl;
  out[4 * i + 1] = ay - dt;
  out[4 * i + 2] = ax + dr;
  out[4 * i + 3] = ay + db;
  out[15000 + i] = (float)label[cid];                // labels_s
}

// ---------------- K7a: parallel IoU suppression bitmask -------------------
// One TDM descriptor stages the whole 48KB box table into LDS per block:
// D# group0 = {count, lds_addr, global_addr, type=2}, group1 = 1-row tile of
// 12000 x 4B elements. TENSORcnt tracked; wave 0 issues, all waves consume.
__global__ __launch_bounds__(256) void k_nms_mask(const float* __restrict__ out,
                                                  uint32_t* __restrict__ mask) {
  __shared__ float sb[NSEL * 4];                     // 48 KB
  int tid = threadIdx.x;
  if (tid < 32) {                                    // wave 0: single TDM DMA
    uint64_t ga = (uint64_t)(uintptr_t)out;          // boxes at out[0..11999]
    u32x4 g0;
    g0[0] = 1u;                                      // count=1, user descriptor
    g0[1] = lds_off(&sb[0]);                         // lds_addr (bytes)
    g0[2] = (uint32_t)ga;                            // global_addr[31:0]
    g0[3] = ((uint32_t)(ga >> 32) & 0x01FFFFFFu)     // global_addr[56:32]
            | (2u << 30);                            // type=2 ("image")
    u32x8 g1;
    g1[0] = 2u << 16;                                // data_size=4B; mask=0; no pad
    g1[1] = (12000u & 0xFFFFu) << 16;                // tensor_dim0[15:0]
    g1[2] = (12000u >> 16) | (1u << 16);             // tensor_dim0[31:16] | tensor_dim1=1
    g1[3] = (12000u & 0xFFFFu) << 16;                // tile_dim0 = 12000
    g1[4] = 1u;                                      // tile_dim1 = 1, tile_dim2 = 0
    g1[5] = 12000u;                                  // tensor_dim0_stride[31:0]
    g1[6] = 0u;                                      // stride hi / dim1_stride lo
    g1[7] = 0u;
    asm volatile("tensor_load_to_lds %0, %1" :: "s"(g0), "s"(g1) : "memory");
    __builtin_amdgcn_s_wait_tensorcnt(0);
  }
  __syncthreads();
  int g = blockIdx.x * 256 + tid;
  if (g >= NSEL * NWORDS) return;
  int i = g / NWORDS, w = g % NWORDS;
  float x1 = sb[4 * i], y1 = sb[4 * i + 1], x2 = sb[4 * i + 2], y2 = sb[4 * i + 3];
  float ai = (x2 - x1) * (y2 - y1);
  uint32_t m = 0;
  int j0 = w * 32;
#pragma unroll 4
  for (int b = 0; b < 32; ++b) {
    int j = j0 + b;
    if (j > i && j < NSEL) {
      float u1 = fmaxf(x1, sb[4 * j]),     v1 = fmaxf(y1, sb[4 * j + 1]);
      float u2 = fminf(x2, sb[4 * j + 2]), v2 = fminf(y2, sb[4 * j + 3]);
      float ww = fmaxf(1e-10f, u2 - u1), hh = fmaxf(1e-10f, v2 - v1);
      float inter = ww * hh;
      float aj = (sb[4 * j + 2] - sb[4 * j]) * (sb[4 * j + 3] - sb[4 * j + 1]);
      float iou = inter / (ai + aj - inter + 1e-14f);
      if (iou > NMS_TH) m |= (1u << b);
    }
  }
  mask[g] = m;
}

// ---------------- K7b: single-wave greedy scan (wave32 lockstep) ----------
// Single-wave workgroup: barrier ops lower to S_NOP, so the serial chain is
// pure LDS latency. keep[i] == !removed[i] at the end of the greedy scan.
__global__ void k_nms_scan(const uint32_t* __restrict__ mask, float* __restrict__ out) {
  __shared__ uint32_t rem[NWORDS];
  int tid = threadIdx.x;                             // blockDim == 32
  for (int w = tid; w < NWORDS; w += 32) rem[w] = 0u;
  __syncthreads();
  for (int i = 0; i < NSEL; ++i) {
    bool alive = ((rem[i >> 5] >> (i & 31)) & 1u) == 0u;
    if (alive) {
      for (int w = tid; w < NWORDS; w += 32)
        rem[w] |= mask[i * NWORDS + w];              // row has only bits j>i
    }
    __syncthreads();
  }
  for (int i = tid; i < NSEL; i += 32) {
    bool keep = (((rem[i >> 5] >> (i & 31)) & 1u) == 0u) && (out[12000 + i] > CONF_TH);
    out[18000 + i] = keep ? 1.0f : 0.0f;             // keep mask
  }
}

// ---------------- host launch ----------------
extern "C" void kernel_launch(void* const* d_in, const int* in_sizes, int n_in,
                              void* d_out, int out_size, void* d_ws, size_t ws_size,
                              hipStream_t stream) {
  (void)in_sizes; (void)n_in; (void)out_size; (void)ws_size;
  // setup_inputs order: cls0, reg0, ctn0, cls1, reg1, ctn1, cls2, reg2, ctn2, scales
  const float* cls0 = (const float*)d_in[0];
  const float* reg0 = (const float*)d_in[1];
  const float* ctn0 = (const float*)d_in[2];
  const float* cls1 = (const float*)d_in[3];
  const float* reg1 = (const float*)d_in[4];
  const float* ctn1 = (const float*)d_in[5];
  const float* cls2 = (const float*)d_in[6];
  const float* reg2 = (const float*)d_in[7];
  const float* ctn2 = (const float*)d_in[8];
  const float* scales = (const float*)d_in[9];
  float* out = (float*)d_out;
  uint32_t* ws = (uint32_t*)d_ws;

  float*    score  = (float*)(ws + OFF_SCORE);
  uint32_t* label  = ws + OFF_LABEL;
  uint32_t* h1     = ws + OFF_H1;
  uint32_t* h2     = ws + OFF_H2;
  uint32_t* meta   = ws + OFF_META;
  uint32_t* cnt    = ws + OFF_CNT;
  uint32_t* selCid = ws + OFF_SELCID;
  uint32_t* selKey = ws + OFF_SELKEY;
  uint32_t* sorted = ws + OFF_SORT;
  uint32_t* mask   = ws + OFF_MASK;

  int gCand = (T_CAND + 255) / 256;
  k_zero<<<(ZERO_N + 255) / 256, 256, 0, stream>>>(ws);
  k_decode<<<(T_CAND / 4 + 255) / 256, 256, 0, stream>>>(cls0, cls1, cls2,
                                                         ctn0, ctn1, ctn2, score, label);
  k_hist1<<<gCand, 256, 0, stream>>>(score, h1);
  k_sel1<<<1, 64, 0, stream>>>(h1, meta);
  k_hist2<<<gCand, 256, 0, stream>>>(score, meta, h2);
  k_sel2<<<1, 64, 0, stream>>>(h2, meta);
  k_compact<<<gCand, 256, 0, stream>>>(score, meta, cnt, selCid, selKey);
  k_sort<<<1, 1024, 0, stream>>>(selKey, selCid, sorted, out);
  k_boxes<<<(NSEL + 255) / 256, 256, 0, stream>>>(sorted, label, reg0, reg1, reg2, scales, out);
  k_nms_mask<<<(NSEL * NWORDS + 255) / 256, 256, 0, stream>>>(out, mask);
  k_nms_scan<<<1, 32, 0, stream>>>(mask, out);
}